// Bilinear_31911607009670
// MI455X (gfx1250) — compile-verified
//
#include <hip/hip_runtime.h>

// Bilinear: out[b,l,m] = t0[l,b,:] @ K @ t1[m,b,:] + bias
//   L=4096, B=4, D=64.
// Stage 1: Q[m,b,d] = sum_e t1[m,b,e] * K[d,e]     (16384x64 @ 64x64)
// Stage 2: out[b,l,m] = sum_d t0[l,b,d] * Q[m,b,d] (per-b 4096x64 @ 64x4096)
// fp32 WMMA path: V_WMMA_F32_16X16X4_F32, 64x64 macro-tile per wave.

typedef __attribute__((ext_vector_type(2))) float v2f;
typedef __attribute__((ext_vector_type(8))) float v8f;

#define L_DIM 4096
#define B_DIM 4
#define D_DIM 64

__device__ __forceinline__ v8f wmma4(v2f a, v2f b, v8f c) {
  // 8 args: (neg_a, A, neg_b, B, c_mod, C, reuse_a, reuse_b)
  return __builtin_amdgcn_wmma_f32_16x16x4_f32(
      /*neg_a=*/false, a, /*neg_b=*/false, b,
      /*c_mod=*/(short)0, c, /*reuse_a=*/false, /*reuse_b=*/false);
}

// -------------------------------------------------------------------------
// Stage 1: Q = T1_flat (R x 64) @ K^T, R = L*B = 16384.
// One wave -> 64 rows x 64 cols (full N). 256 waves total.
// A frag: A[idx][k0+2h+v] = t1[row*64 + ...]          (contiguous float2)
// B frag: (K^T)[k][n] = K[n][k] -> K[(c*16+idx)*64 + k0+2h+v] (contiguous)
// -------------------------------------------------------------------------
__global__ __launch_bounds__(256) void bilinear_stage1(
    const float* __restrict__ t1, const float* __restrict__ kern,
    float* __restrict__ q) {
  const int wid  = blockIdx.x * (blockDim.x >> 5) + (threadIdx.x >> 5);
  const int lane = threadIdx.x & 31;
  const int half = lane >> 4;
  const int idx  = lane & 15;
  const int r0   = wid * 64;

  v8f acc[4][4];
#pragma unroll
  for (int r = 0; r < 4; ++r)
#pragma unroll
    for (int c = 0; c < 4; ++c)
#pragma unroll
      for (int j = 0; j < 8; ++j) acc[r][c][j] = 0.0f;

  const float* arow[4];
  const float* brow[4];
#pragma unroll
  for (int r = 0; r < 4; ++r)
    arow[r] = t1 + (size_t)(r0 + r * 16 + idx) * D_DIM;
#pragma unroll
  for (int c = 0; c < 4; ++c)
    brow[c] = kern + (size_t)(c * 16 + idx) * D_DIM;

#pragma unroll
  for (int k0 = 0; k0 < D_DIM; k0 += 4) {
    v2f a[4], bf[4];
#pragma unroll
    for (int r = 0; r < 4; ++r)
      a[r] = *(const v2f*)(arow[r] + k0 + 2 * half);
#pragma unroll
    for (int c = 0; c < 4; ++c)
      bf[c] = *(const v2f*)(brow[c] + k0 + 2 * half);
#pragma unroll
    for (int r = 0; r < 4; ++r)
#pragma unroll
      for (int c = 0; c < 4; ++c)
        acc[r][c] = wmma4(a[r], bf[c], acc[r][c]);
  }

#pragma unroll
  for (int r = 0; r < 4; ++r) {
#pragma unroll
    for (int j = 0; j < 8; ++j) {
      const int row = r0 + r * 16 + j + 8 * half;
      float* qrow = q + (size_t)row * D_DIM;
#pragma unroll
      for (int c = 0; c < 4; ++c)
        qrow[c * 16 + idx] = acc[r][c][j];
    }
  }
}

// -------------------------------------------------------------------------
// Stage 2: per batch b, out[b] = T0_b @ Q_b^T.
// One wave -> one 64x64 output macro-tile. 4*64*64 = 16384 waves.
// A frag from t0 rows [l0+r*16+idx, b, :], B frag from Q rows
// [m0+c*16+idx, b, :] — both contiguous float2 along d.
// -------------------------------------------------------------------------
__global__ __launch_bounds__(256) void bilinear_stage2(
    const float* __restrict__ t0, const float* __restrict__ q,
    const float* __restrict__ bias, float* __restrict__ out) {
  const int wid  = blockIdx.x * (blockDim.x >> 5) + (threadIdx.x >> 5);
  const int lane = threadIdx.x & 31;
  const int half = lane >> 4;
  const int idx  = lane & 15;

  const int tilesPerBatch = (L_DIM / 64) * (L_DIM / 64);  // 4096
  const int b   = wid / tilesPerBatch;
  const int rem = wid - b * tilesPerBatch;
  const int tm  = rem >> 6;
  const int tn  = rem & 63;
  const int l0  = tm * 64;
  const int m0  = tn * 64;

  v8f acc[4][4];
#pragma unroll
  for (int r = 0; r < 4; ++r)
#pragma unroll
    for (int c = 0; c < 4; ++c)
#pragma unroll
      for (int j = 0; j < 8; ++j) acc[r][c][j] = 0.0f;

  const float* arow[4];
  const float* brow[4];
#pragma unroll
  for (int r = 0; r < 4; ++r)
    arow[r] = t0 + ((size_t)(l0 + r * 16 + idx) * B_DIM + b) * D_DIM;
#pragma unroll
  for (int c = 0; c < 4; ++c)
    brow[c] = q + ((size_t)(m0 + c * 16 + idx) * B_DIM + b) * D_DIM;

#pragma unroll
  for (int k0 = 0; k0 < D_DIM; k0 += 4) {
    v2f a[4], bf[4];
#pragma unroll
    for (int r = 0; r < 4; ++r)
      a[r] = *(const v2f*)(arow[r] + k0 + 2 * half);
#pragma unroll
    for (int c = 0; c < 4; ++c)
      bf[c] = *(const v2f*)(brow[c] + k0 + 2 * half);
#pragma unroll
    for (int r = 0; r < 4; ++r)
#pragma unroll
      for (int c = 0; c < 4; ++c)
        acc[r][c] = wmma4(a[r], bf[c], acc[r][c]);
  }

  const float bv = bias[0];
  const size_t outBase = (size_t)b * L_DIM * L_DIM;
#pragma unroll
  for (int r = 0; r < 4; ++r) {
#pragma unroll
    for (int j = 0; j < 8; ++j) {
      const int l = l0 + r * 16 + j + 8 * half;
      float* orow = out + outBase + (size_t)l * L_DIM + m0;
#pragma unroll
      for (int c = 0; c < 4; ++c)
        __builtin_nontemporal_store(acc[r][c][j] + bv, orow + c * 16 + idx);
    }
  }
}

extern "C" void kernel_launch(void* const* d_in, const int* in_sizes, int n_in,
                              void* d_out, int out_size, void* d_ws, size_t ws_size,
                              hipStream_t stream) {
  const float* t0   = (const float*)d_in[0];  // [L, B, D]
  const float* t1   = (const float*)d_in[1];  // [L, B, D]
  const float* kern = (const float*)d_in[2];  // [D, D]
  const float* bias = (const float*)d_in[3];  // [1]
  float* out = (float*)d_out;                 // [1, B, L, L]
  float* q   = (float*)d_ws;                  // needs L*B*D*4 = 4 MB

  // Stage 1: R = L*B = 16384 rows, 64 rows per wave -> 256 waves, 8 waves/block.
  const int s1_waves  = (L_DIM * B_DIM) / 64;       // 256
  const int s1_blocks = s1_waves / 8;               // 32
  hipLaunchKernelGGL(bilinear_stage1, dim3(s1_blocks), dim3(256), 0, stream,
                     t1, kern, q);

  // Stage 2: B * (L/64)^2 = 16384 macro-tiles, 1 wave each, 8 waves/block.
  const int s2_waves  = B_DIM * (L_DIM / 64) * (L_DIM / 64);  // 16384
  const int s2_blocks = s2_waves / 8;                         // 2048
  hipLaunchKernelGGL(bilinear_stage2, dim3(s2_blocks), dim3(256), 0, stream,
                     t0, q, bias, out);
}